// smoothTransformer2D_83614423318532
// MI455X (gfx1250) — compile-verified
//
#include <hip/hip_runtime.h>
#include <hip/hip_bf16.h>

#define BB 8
#define HH 512
#define WW 512
#define CC 16

typedef float v2f __attribute__((ext_vector_type(2)));
typedef float v4f __attribute__((ext_vector_type(4)));
typedef float v8f __attribute__((ext_vector_type(8)));

__device__ __forceinline__ float dg_sigmoid(float x) {
    // dg = c / (1 + (c-1)*exp(-x)), c = MAXGRAD = 2
    return 2.0f / (1.0f + expf(-x));
}

// One wave = one 512-element sequence (a row for the x-pass, a column for the
// y-pass). Inclusive scan done as triangular-ones matmul with
// V_WMMA_F32_16X16X4_F32: D[m,j] = sum_{k<=m} B[k,j], i.e. each of 16 columns
// is a 16-element segment scan; two 16x16 matmuls (4 chained K=4 WMMAs each)
// cover 512 elements; segment totals are combined with a 32-lane shuffle scan.
__global__ __launch_bounds__(256) void scan512_grid_kernel(
    const float* __restrict__ dg, float* __restrict__ gnorm,
    float* __restrict__ ginv, int isY)
{
    const int lane = threadIdx.x & 31;
    const int wave = threadIdx.x >> 5;
    const int s = blockIdx.x * 8 + wave;          // sequence id, 0..4095

    int base, stride;
    if (isY) {  // column (b, w): element e = defgrad[b, e, w, 1]
        base = (s >> 9) * (HH * WW * 2) + (s & 511) * 2 + 1;
        stride = WW * 2;
    } else {    // row (b, h): element e = defgrad[b, h, e, 0]
        base = s * (WW * 2);
        stride = 2;
    }

    const int j    = lane & 15;   // column index (= segment within matrix)
    const int half = lane >> 4;   // lanes 16-31 hold the "other" K/M halves

#if __has_builtin(__builtin_amdgcn_wmma_f32_16x16x4_f32)
    v8f d0 = {0.f,0.f,0.f,0.f,0.f,0.f,0.f,0.f};   // scans of elements   0..255
    v8f d1 = {0.f,0.f,0.f,0.f,0.f,0.f,0.f,0.f};   // scans of elements 256..511
#pragma unroll
    for (int sl = 0; sl < 4; ++sl) {              // K slabs: k = 4*sl .. 4*sl+3
        const int k0 = 4 * sl + 2 * half;
        // A (16x4 slab of lower-triangular ones): A[m,k] = (k <= m)
        v2f a;
        a.x = (k0     <= j) ? 1.0f : 0.0f;
        a.y = (k0 + 1 <= j) ? 1.0f : 0.0f;
        // B (4x16): B[k,j] = value of element j*16+k
        const int e0 = j * 16 + k0;
        v2f b0, b1;
        b0.x = dg_sigmoid(dg[base + (e0    ) * stride]);
        b0.y = dg_sigmoid(dg[base + (e0 + 1) * stride]);
        b1.x = dg_sigmoid(dg[base + (e0 + 256) * stride]);
        b1.y = dg_sigmoid(dg[base + (e0 + 257) * stride]);
        d0 = __builtin_amdgcn_wmma_f32_16x16x4_f32(false, a, false, b0,
                                                   (short)0, d0, false, false);
        d1 = __builtin_amdgcn_wmma_f32_16x16x4_f32(false, a, false, b1,
                                                   (short)0, d1, false, false);
    }

    // Segment totals: D[15,j] lives in VGPR7, lanes 16..31.
    float t0 = __shfl(d0[7], 16 + j, 32);         // total of segment j, matrix 0
    float tot = (half == 0) ? t0 : d1[7];         // lane g holds total of seg g

    // Inclusive 32-lane scan of segment totals, then exclusive offsets.
    float sc = tot;
#pragma unroll
    for (int off = 1; off < 32; off <<= 1) {
        float n = __shfl_up(sc, off, 32);
        if (lane >= off) sc += n;
    }
    float ex = __shfl_up(sc, 1, 32);
    if (lane == 0) ex = 0.0f;
    const float off0  = __shfl(ex, j, 32);
    const float off1  = __shfl(ex, 16 + j, 32);
    const float first = __shfl(d0[0], 0, 32);     // scan[0]   (offset is 0)
    const float last  = __shfl(sc, 31, 32);       // scan[511] (grand total)
    const float scl   = 511.0f / (last - first);

#pragma unroll
    for (int v = 0; v < 8; ++v) {
        const int mr = v + 8 * half;              // matrix row held in VGPR v
        int e = j * 16 + mr;
        float r   = d0[v] + off0;
        float nrm = (r - first) * scl;
        gnorm[base + e * stride] = nrm;
        ginv [base + e * stride] = 2.0f * (float)e - nrm;
        e += 256;
        r   = d1[v] + off1;
        nrm = (r - first) * scl;
        gnorm[base + e * stride] = nrm;
        ginv [base + e * stride] = 2.0f * (float)e - nrm;
    }
#else
    // VALU fallback: lane g serially scans segment g (16 elements).
    float vals[16];
    float acc = 0.0f;
#pragma unroll
    for (int k = 0; k < 16; ++k) {
        acc += dg_sigmoid(dg[base + (lane * 16 + k) * stride]);
        vals[k] = acc;
    }
    float sc = acc;
#pragma unroll
    for (int off = 1; off < 32; off <<= 1) {
        float n = __shfl_up(sc, off, 32);
        if (lane >= off) sc += n;
    }
    float ex = __shfl_up(sc, 1, 32);
    if (lane == 0) ex = 0.0f;
    const float first = __shfl(vals[0], 0, 32);
    const float last  = __shfl(sc, 31, 32);
    const float scl   = 511.0f / (last - first);
#pragma unroll
    for (int k = 0; k < 16; ++k) {
        const int e = lane * 16 + k;
        float nrm = (vals[k] + ex - first) * scl;
        gnorm[base + e * stride] = nrm;
        ginv [base + e * stride] = 2.0f * (float)e - nrm;
    }
#endif
}

// Bilinear resample of a (B,512,512,16) f32 image. 4 lanes per pixel, each
// lane handles a float4 channel quad -> pure B128 traffic. NT=true streams
// the output with non-temporal stores so it does not evict mov_def from the
// 192MB L2 (mov_def is re-read by the second resample pass).
template <bool NT>
__global__ __launch_bounds__(256) void resample16_kernel(
    const float* __restrict__ im, const float* __restrict__ grid,
    float* __restrict__ out)
{
    const int t = blockIdx.x * 256 + threadIdx.x;
    const int p = t >> 2;            // pixel index within (B,H,W)
    const int q = t & 3;             // channel quad
    const int b = p >> 18;           // H*W = 2^18

    const float gx = grid[2 * p];
    const float gy = grid[2 * p + 1];

    int x0 = (int)floorf(gx), y0 = (int)floorf(gy);
    int x1 = x0 + 1, y1 = y0 + 1;
    x0 = min(max(x0, 0), WW - 1);
    x1 = min(max(x1, 0), WW - 1);
    y0 = min(max(y0, 0), HH - 1);
    y1 = min(max(y1, 0), HH - 1);
    const float x0f = (float)x0, x1f = (float)x1;
    const float y0f = (float)y0, y1f = (float)y1;
    const float wa = (x1f - gx) * (y1f - gy);
    const float wb = (x1f - gx) * (gy  - y0f);
    const float wc = (gx  - x0f) * (y1f - gy);
    const float wd = (gx  - x0f) * (gy  - y0f);

    const int bb = b << 18;          // b*H*W
    const int c4 = q << 2;
    const v4f Ia = *(const v4f*)(im + (((bb + (y0 << 9) + x0) << 4) + c4));
    const v4f Ib = *(const v4f*)(im + (((bb + (y1 << 9) + x0) << 4) + c4));
    const v4f Ic = *(const v4f*)(im + (((bb + (y0 << 9) + x1) << 4) + c4));
    const v4f Id = *(const v4f*)(im + (((bb + (y1 << 9) + x1) << 4) + c4));

    v4f r = wa * Ia + wb * Ib + wc * Ic + wd * Id;
    float* dst = out + ((p << 4) + c4);
    if (NT) __builtin_nontemporal_store(r, (v4f*)dst);
    else    *(v4f*)dst = r;
}

extern "C" void kernel_launch(void* const* d_in, const int* in_sizes, int n_in,
                              void* d_out, int out_size, void* d_ws, size_t ws_size,
                              hipStream_t stream) {
    (void)in_sizes; (void)n_in; (void)out_size; (void)d_ws; (void)ws_size;
    const float* mov     = (const float*)d_in[0];
    // d_in[1] ("ref") is unused by the reference computation.
    const float* defgrad = (const float*)d_in[2];

    const size_t N_IM = (size_t)BB * HH * WW * CC;   // 33,554,432
    const size_t N_G  = (size_t)BB * HH * WW * 2;    //  4,194,304

    float* mov_def = (float*)d_out;                  // output 0
    float* ref_def = mov_def + N_IM;                 // output 1
    float* gnorm   = mov_def + 2 * N_IM;             // output 2
    float* ginv    = gnorm + N_G;                    // output 3

    // Grid passes: 4096 sequences each, 1 wave per sequence, 8 waves/block.
    scan512_grid_kernel<<<512, 256, 0, stream>>>(defgrad, gnorm, ginv, 0);
    scan512_grid_kernel<<<512, 256, 0, stream>>>(defgrad, gnorm, ginv, 1);

    // Resample passes: 4 lanes per pixel.
    const int nthreads = BB * HH * WW * 4;           // 8,388,608
    resample16_kernel<false><<<nthreads / 256, 256, 0, stream>>>(mov, gnorm, mov_def);
    resample16_kernel<true ><<<nthreads / 256, 256, 0, stream>>>(mov_def, ginv, ref_def);
}